// MemoryAccess_1365799600260
// MI455X (gfx1250) — compile-verified
//
#include <hip/hip_runtime.h>
#include <math.h>

typedef __attribute__((ext_vector_type(2))) float v2f;
typedef __attribute__((ext_vector_type(8))) float v8f;

// Problem constants
constexpr int Dd = 1024;          // controller dim
constexpr float EPSf = 1e-6f;

// Workspace layout (floats)
constexpr size_t WS_WV = 0;         // 1024*64  raw write vectors
constexpr size_t WS_EV = 65536;     // 1024*64  sigmoid erase vectors
constexpr size_t WS_FG = 131072;    // 1024*4   sigmoid free gate
constexpr size_t WS_AG = 135168;    // 1024*1   sigmoid alloc gate
constexpr size_t WS_WG = 136192;    // 1024*1   sigmoid write gate
constexpr size_t WS_RM = 137216;    // 1024*12  raw read modes
constexpr size_t WS_WSP = 149504;   // 1024*1   softplus write strength
constexpr size_t WS_RS = 150528;    // 1024*4   softplus read strengths
constexpr size_t WS_WK = 154624;    // 1024*64  write keys
constexpr size_t WS_RK = 220160;    // 1024*256 read keys
constexpr size_t WS_FW = 482304;    // 1024*4*256 forward_w
constexpr size_t WS_BW = 1530880;   // 1024*4*256 backward_w

// Output layout (floats), reference tuple order
constexpr size_t O_RWORDS = 0;           // (B,R,WD)   262144
constexpr size_t O_MEM    = 262144;      // (B,N,WD)   16777216
constexpr size_t O_RWTS   = 17039360;    // (B,R,N)    1048576
constexpr size_t O_WW     = 18087936;    // (B,NW,N)   262144
constexpr size_t O_LINK   = 18350080;    // (B,NW,N,N) 67108864
constexpr size_t O_PREC   = 85458944;    // (B,NW,N)   262144
constexpr size_t O_USAGE  = 85721088;    // (B,N)      262144

// ---------------- block reduction helpers (256 threads) ----------------
__device__ __forceinline__ float blk_max(float* red, int t, float v) {
    red[t] = v; __syncthreads();
    #pragma unroll
    for (int s = 128; s > 0; s >>= 1) {
        if (t < s) red[t] = fmaxf(red[t], red[t + s]);
        __syncthreads();
    }
    float r = red[0]; __syncthreads(); return r;
}
__device__ __forceinline__ float blk_sum(float* red, int t, float v) {
    red[t] = v; __syncthreads();
    #pragma unroll
    for (int s = 128; s > 0; s >>= 1) {
        if (t < s) red[t] = red[t] + red[t + s];
        __syncthreads();
    }
    float r = red[0]; __syncthreads(); return r;
}

// ---------------- Kernel A: WMMA f32 GEMM  Y = ctrl @ W + b, + act ------
// One wave32 per 16x16 output tile, K-loop in steps of 4 using
// V_WMMA_F32_16X16X4_F32.  Out-of-range columns handled branch-free via a
// clamped index + 0/1 mask so the inner loop has NO exec-mask churn.
// ACT: 0=none, 1=sigmoid, 2=softplus.
template <int OD, int ACT>
__global__ void __launch_bounds__(32) k_proj_wmma(
    const float* __restrict__ A, const float* __restrict__ W,
    const float* __restrict__ bias, float* __restrict__ Y)
{
    const int lane = threadIdx.x;
    const int half = lane >> 4, l = lane & 15;
    const int m0 = blockIdx.x * 16, n0 = blockIdx.y * 16;
    const int col = n0 + l;
    const bool cok = (OD % 16 == 0) ? true : (col < OD);
    const int ccol = cok ? col : 0;            // always-valid address
    const float msk = cok ? 1.f : 0.f;         // zero out-of-range columns
    const float* arow = A + (size_t)(m0 + l) * Dd + 2 * half;
    const float* wcol = W + ccol + (size_t)(2 * half) * OD;

    v8f acc = {};
    #pragma unroll 4
    for (int k = 0; k < Dd; k += 4) {
        // A fragment: lanes 0-15 -> K=k,k+1 ; lanes 16-31 -> K=k+2,k+3
        v2f a;
        { float2 t2 = *(const float2*)(arow + k); a.x = t2.x; a.y = t2.y; }
        // B fragment: lanes 0-15 -> K=k,k+1 ; lanes 16-31 -> K=k+2,k+3 at col
        v2f bf;
        bf.x = wcol[(size_t)k * OD] * msk;
        bf.y = wcol[(size_t)(k + 1) * OD] * msk;
        acc = __builtin_amdgcn_wmma_f32_16x16x4_f32(
            false, a, false, bf, (short)0, acc, false, false);
    }
    if (cok) {
        const float bv = bias[col];
        #pragma unroll
        for (int i = 0; i < 8; i++) {
            const int row = m0 + i + 8 * half;  // C/D layout: VGPR i -> M=i / i+8
            float v = acc[i] + bv;
            if (ACT == 1)      v = 1.f / (1.f + expf(-v));
            else if (ACT == 2) v = (v > 20.f) ? v : log1pf(expf(v));
            Y[(size_t)row * OD + col] = v;
        }
    }
}

// ---------------- Kernel B: per-batch usage / content / allocation ------
__global__ void __launch_bounds__(256) k_write_stage(
    const float* __restrict__ mem, const float* __restrict__ prw,
    const float* __restrict__ pww, const float* __restrict__ pprec,
    const float* __restrict__ pusage, const float* __restrict__ ws,
    float* __restrict__ out)
{
    __shared__ float s_usage[256], s_cont[256], skey[256], s_excl[256],
                     s_alloc[256], s_red[256], s_wk[64];
    __shared__ float s_kn;
    __shared__ int sidx[256];
    const int b = blockIdx.x, t = threadIdx.x;

    // usage = (pu + (1-pu)*agg_ww) * phi
    const float pu  = pusage[(size_t)b * 256 + t];
    const float agg = pww[(size_t)b * 256 + t];      // NW == 1
    const float uaw = pu + (1.f - pu) * agg;
    float phi = 1.f;
    #pragma unroll
    for (int r = 0; r < 4; r++)
        phi *= 1.f - ws[WS_FG + (size_t)b * 4 + r] * prw[(size_t)b * 1024 + r * 256 + t];
    const float usage = uaw * phi;
    s_usage[t] = usage;
    out[O_USAGE + (size_t)b * 256 + t] = usage;

    if (t < 64) s_wk[t] = ws[WS_WK + (size_t)b * 64 + t];
    __syncthreads();
    if (t == 0) {
        float s = 0.f;
        for (int d = 0; d < 64; d++) s += s_wk[d] * s_wk[d];
        s_kn = sqrtf(s);
    }
    __syncthreads();

    // write content: cosine sim vs old memory, 4 threads per row (coalesced)
    const float strength = ws[WS_WSP + b];
    for (int c = 0; c < 4; c++) {
        const int n = c * 64 + (t >> 2), q = t & 3;
        const float* mrow = mem + ((size_t)b * 256 + n) * 64 + q * 16;
        float mn2 = 0.f, dot = 0.f;
        #pragma unroll
        for (int dd = 0; dd < 16; dd++) {
            const float mv = mrow[dd];
            mn2 += mv * mv;
            dot += mv * s_wk[q * 16 + dd];
        }
        mn2 += __shfl_xor(mn2, 1); mn2 += __shfl_xor(mn2, 2);
        dot += __shfl_xor(dot, 1); dot += __shfl_xor(dot, 2);
        if (q == 0) s_cont[n] = dot / (s_kn * sqrtf(mn2) + EPSf) * strength;
    }
    __syncthreads();
    {   // softmax over N
        const float v = s_cont[t];
        const float m = blk_max(s_red, t, v);
        const float e = expf(v - m);
        const float Z = blk_sum(s_red, t, e);
        s_cont[t] = e / Z;
    }
    __syncthreads();

    // allocation: sort nonusage descending (bitonic), exclusive cumprod
    const float up = EPSf + (1.f - EPSf) * s_usage[t];
    skey[t] = 1.f - up; sidx[t] = t;
    __syncthreads();
    for (int k = 2; k <= 256; k <<= 1)
        for (int j = k >> 1; j > 0; j >>= 1) {
            const int ixj = t ^ j;
            if (ixj > t) {
                const bool desc = ((t & k) == 0);
                const float kt = skey[t], kx = skey[ixj];
                if (desc ? (kt < kx) : (kt > kx)) {
                    skey[t] = kx; skey[ixj] = kt;
                    const int it = sidx[t]; sidx[t] = sidx[ixj]; sidx[ixj] = it;
                }
            }
            __syncthreads();
        }
    if (t == 0) {
        float p = 1.f;
        for (int pos = 0; pos < 256; pos++) { s_excl[pos] = p; p *= (1.f - skey[pos]); }
    }
    __syncthreads();
    s_alloc[sidx[t]] = skey[t] * s_excl[t];
    __syncthreads();

    // write weights + precedence
    const float agv = ws[WS_AG + b], wgv = ws[WS_WG + b];
    const float wwv = wgv * (agv * s_alloc[t] + (1.f - agv) * s_cont[t]);
    out[O_WW + (size_t)b * 256 + t] = wwv;
    const float sww = blk_sum(s_red, t, wwv);
    out[O_PREC + (size_t)b * 256 + t] =
        (1.f - sww) * pprec[(size_t)b * 256 + t] + wwv;
}

// ---------------- Kernel C: memory erase + write (float4 streaming) -----
__global__ void __launch_bounds__(256) k_mem_update(
    const float* __restrict__ mem, const float* __restrict__ ww,
    const float* __restrict__ ev, const float* __restrict__ wv,
    float* __restrict__ memo)
{
    const size_t idx = (size_t)blockIdx.x * 256 + threadIdx.x;  // float4 id
    const int d4 = (int)(idx & 15);
    const int n  = (int)((idx >> 4) & 255);
    const int b  = (int)(idx >> 12);
    const float w = ww[(size_t)b * 256 + n];
    const float4 m = *(const float4*)(mem + idx * 4);
    const float4 e = *(const float4*)(ev + (size_t)b * 64 + d4 * 4);
    const float4 v = *(const float4*)(wv + (size_t)b * 64 + d4 * 4);
    float4 o;
    o.x = m.x * (1.f - w * e.x) + w * v.x;
    o.y = m.y * (1.f - w * e.y) + w * v.y;
    o.z = m.z * (1.f - w * e.z) + w * v.z;
    o.w = m.w * (1.f - w * e.w) + w * v.w;
    *(float4*)(memo + idx * 4) = o;
}

// ---------------- Kernel D: fused link + forward_w + backward_w ---------
// One block per batch. Thread t owns column j=t (coalesced link stream).
// 32-row LDS tiles provide the row-direction (forward_w) reduction.
__global__ void __launch_bounds__(256) k_link_stage(
    const float* __restrict__ pl, const float* __restrict__ prw,
    const float* __restrict__ pprec, const float* __restrict__ ww,
    float* __restrict__ link, float* __restrict__ fw, float* __restrict__ bw)
{
    __shared__ float s_ww[256], s_pp[256], s_prw[1024], s_fw[1024];
    __shared__ float sL[32 * 260];
    const int b = blockIdx.x, t = threadIdx.x;
    s_ww[t] = ww[(size_t)b * 256 + t];
    s_pp[t] = pprec[(size_t)b * 256 + t];
    #pragma unroll
    for (int r = 0; r < 4; r++) {
        s_prw[r * 256 + t] = prw[(size_t)b * 1024 + r * 256 + t];
        s_fw[r * 256 + t] = 0.f;
    }
    __syncthreads();

    const float* plb = pl + (size_t)b * 65536;
    float* lkb = link + (size_t)b * 65536;
    const float wwt = s_ww[t], ppt = s_pp[t];
    float bw0 = 0.f, bw1 = 0.f, bw2 = 0.f, bw3 = 0.f;

    for (int ib = 0; ib < 8; ib++) {
        if (ib + 1 < 8)                         // gfx1250 global_prefetch_b8
            __builtin_prefetch(plb + (ib + 1) * 8192 + t * 32, 0, 0);
        for (int ii = 0; ii < 32; ii++) {
            const int i = ib * 32 + ii;
            const float wwi = s_ww[i];
            float v = (1.f - wwi - wwt) * plb[i * 256 + t] + wwi * ppt;
            if (i == t) v = 0.f;                // zero diagonal
            lkb[i * 256 + t] = v;
            sL[ii * 260 + t] = v;
            bw0 += s_prw[i] * v;                // bw[r,j] += prw[r,i]*L[i,j]
            bw1 += s_prw[256 + i] * v;
            bw2 += s_prw[512 + i] * v;
            bw3 += s_prw[768 + i] * v;
        }
        __syncthreads();
        {   // fw[r,i] = sum_j prw[r,j]*L[i,j] ; 2 threads per (r,row)
            const int rl = t & 31, r = (t >> 5) & 3, hf = t >> 7;
            const float* pr = s_prw + r * 256 + hf * 128;
            const float* Lr = sL + rl * 260 + hf * 128;
            float sum = 0.f;
            for (int jj = 0; jj < 128; jj++) sum += pr[jj] * Lr[jj];
            atomicAdd(&s_fw[r * 256 + ib * 32 + rl], sum);
        }
        __syncthreads();
    }
    #pragma unroll
    for (int r = 0; r < 4; r++)
        fw[(size_t)b * 1024 + r * 256 + t] = s_fw[r * 256 + t];
    bw[(size_t)b * 1024 + t]        = bw0;
    bw[(size_t)b * 1024 + 256 + t]  = bw1;
    bw[(size_t)b * 1024 + 512 + t]  = bw2;
    bw[(size_t)b * 1024 + 768 + t]  = bw3;
}

// ---------------- Kernel E: read content / weights / words --------------
__global__ void __launch_bounds__(256) k_read_stage(
    const float* __restrict__ memo, const float* __restrict__ ws,
    float* __restrict__ out)
{
    __shared__ float s_cont[1024], s_rw[1024], s_rk[256], s_red[256];
    __shared__ float s_kn[4], s_rm[12];
    const int b = blockIdx.x, t = threadIdx.x;

    s_rk[t] = ws[WS_RK + (size_t)b * 256 + t];
    if (t < 12) s_rm[t] = ws[WS_RM + (size_t)b * 12 + t];
    __syncthreads();
    if (t < 4) {
        float s = 0.f;
        for (int d = 0; d < 64; d++) { const float v = s_rk[t * 64 + d]; s += v * v; }
        s_kn[t] = sqrtf(s);
    }
    float rsv[4];
    #pragma unroll
    for (int r = 0; r < 4; r++) rsv[r] = ws[WS_RS + (size_t)b * 4 + r];
    __syncthreads();

    // read content vs NEW memory (coalesced, 4 threads per row)
    for (int c = 0; c < 4; c++) {
        const int n = c * 64 + (t >> 2), q = t & 3;
        const float* mrow = memo + ((size_t)b * 256 + n) * 64 + q * 16;
        float mn2 = 0.f, d0 = 0.f, d1 = 0.f, d2 = 0.f, d3 = 0.f;
        #pragma unroll
        for (int dd = 0; dd < 16; dd++) {
            const float mv = mrow[dd];
            const int di = q * 16 + dd;
            mn2 += mv * mv;
            d0 += mv * s_rk[di];       d1 += mv * s_rk[64 + di];
            d2 += mv * s_rk[128 + di]; d3 += mv * s_rk[192 + di];
        }
        mn2 += __shfl_xor(mn2, 1); mn2 += __shfl_xor(mn2, 2);
        d0 += __shfl_xor(d0, 1); d0 += __shfl_xor(d0, 2);
        d1 += __shfl_xor(d1, 1); d1 += __shfl_xor(d1, 2);
        d2 += __shfl_xor(d2, 1); d2 += __shfl_xor(d2, 2);
        d3 += __shfl_xor(d3, 1); d3 += __shfl_xor(d3, 2);
        if (q == 0) {
            const float mn = sqrtf(mn2);
            s_cont[n]       = d0 / (s_kn[0] * mn + EPSf) * rsv[0];
            s_cont[256 + n] = d1 / (s_kn[1] * mn + EPSf) * rsv[1];
            s_cont[512 + n] = d2 / (s_kn[2] * mn + EPSf) * rsv[2];
            s_cont[768 + n] = d3 / (s_kn[3] * mn + EPSf) * rsv[3];
        }
    }
    __syncthreads();
    for (int r = 0; r < 4; r++) {
        const float v = s_cont[r * 256 + t];
        const float m = blk_max(s_red, t, v);
        const float e = expf(v - m);
        const float Z = blk_sum(s_red, t, e);
        s_cont[r * 256 + t] = e / Z;
    }
    __syncthreads();

    // read modes softmax (3-way per r) + combine
    const float* fwb = ws + WS_FW + (size_t)b * 1024;
    const float* bwb = ws + WS_BW + (size_t)b * 1024;
    #pragma unroll
    for (int r = 0; r < 4; r++) {
        const float l0 = s_rm[r * 3], l1 = s_rm[r * 3 + 1], l2 = s_rm[r * 3 + 2];
        const float mx = fmaxf(l0, fmaxf(l1, l2));
        const float e0 = expf(l0 - mx), e1 = expf(l1 - mx), e2 = expf(l2 - mx);
        const float iZ = 1.f / (e0 + e1 + e2);
        const float bm = e0 * iZ, fm = e1 * iZ, cm = e2 * iZ;
        const float v = cm * s_cont[r * 256 + t]
                      + fm * fwb[r * 256 + t] + bm * bwb[r * 256 + t];
        s_rw[r * 256 + t] = v;
        out[O_RWTS + (size_t)b * 1024 + r * 256 + t] = v;
    }
    __syncthreads();

    // read words: (4x64) = rw (4x256) @ memory (256x64)
    const int r = t >> 6, d = t & 63;
    const float* mb = memo + (size_t)b * 16384;
    float acc = 0.f;
    for (int n = 0; n < 256; n++) acc += s_rw[r * 256 + n] * mb[n * 64 + d];
    out[O_RWORDS + (size_t)b * 256 + r * 64 + d] = acc;
}

extern "C" void kernel_launch(void* const* d_in, const int* in_sizes, int n_in,
                              void* d_out, int out_size, void* d_ws, size_t ws_size,
                              hipStream_t stream)
{
    (void)in_sizes; (void)n_in; (void)out_size; (void)ws_size;
    const float* ctrl  = (const float*)d_in[0];
    const float* mem   = (const float*)d_in[1];
    const float* prw   = (const float*)d_in[2];
    const float* pww   = (const float*)d_in[3];
    const float* pl    = (const float*)d_in[4];
    const float* pprec = (const float*)d_in[5];
    const float* pus   = (const float*)d_in[6];
    float* ws  = (float*)d_ws;
    float* out = (float*)d_out;

    auto Wp = [&](int i) { return (const float*)d_in[i]; };

    // projections: (weight idx, ws offset, OD, ACT)
    k_proj_wmma<64, 0><<<dim3(64, 4), 32, 0, stream>>>(ctrl, Wp(7),  Wp(8),  ws + WS_WV);
    k_proj_wmma<64, 1><<<dim3(64, 4), 32, 0, stream>>>(ctrl, Wp(9),  Wp(10), ws + WS_EV);
    k_proj_wmma<4,  1><<<dim3(64, 1), 32, 0, stream>>>(ctrl, Wp(11), Wp(12), ws + WS_FG);
    k_proj_wmma<1,  1><<<dim3(64, 1), 32, 0, stream>>>(ctrl, Wp(13), Wp(14), ws + WS_AG);
    k_proj_wmma<1,  1><<<dim3(64, 1), 32, 0, stream>>>(ctrl, Wp(15), Wp(16), ws + WS_WG);
    k_proj_wmma<12, 0><<<dim3(64, 1), 32, 0, stream>>>(ctrl, Wp(17), Wp(18), ws + WS_RM);
    k_proj_wmma<1,  2><<<dim3(64, 1), 32, 0, stream>>>(ctrl, Wp(19), Wp(20), ws + WS_WSP);
    k_proj_wmma<4,  2><<<dim3(64, 1), 32, 0, stream>>>(ctrl, Wp(21), Wp(22), ws + WS_RS);
    k_proj_wmma<64, 0><<<dim3(64, 4), 32, 0, stream>>>(ctrl, Wp(23), Wp(24), ws + WS_WK);
    k_proj_wmma<256,0><<<dim3(64,16), 32, 0, stream>>>(ctrl, Wp(25), Wp(26), ws + WS_RK);

    k_write_stage<<<1024, 256, 0, stream>>>(mem, prw, pww, pprec, pus, ws, out);
    k_mem_update<<<16384, 256, 0, stream>>>(mem, out + O_WW, ws + WS_EV,
                                            ws + WS_WV, out + O_MEM);
    k_link_stage<<<1024, 256, 0, stream>>>(pl, prw, pprec, out + O_WW,
                                           out + O_LINK, ws + WS_FW, ws + WS_BW);
    k_read_stage<<<1024, 256, 0, stream>>>(out + O_MEM, ws, out);
}